// CircuitModel_53188874993954
// MI455X (gfx1250) — compile-verified
//
#include <hip/hip_runtime.h>

// Oja-rule scan:  per (batch, output-row):  w <- s*w + (lr*y)*x,  y = sigmoid(w.x)
// Factored as W_t = P_t * V_t  (P = running product of per-row decays s),
// so the big-matrix update V += coef * (y (x) x) is a pure rank-1 WMMA accumulate.

typedef float v2f __attribute__((ext_vector_type(2)));
typedef float v8f __attribute__((ext_vector_type(8)));

// group-of-32 ds_swizzle: offset[15]=0, xor=offset[14:10], and=offset[4:0]
#define SWZ_XOR_ADD(v, pat)                                                    \
  (v) += __int_as_float(__builtin_amdgcn_ds_swizzle(__float_as_int(v), (pat)))

constexpr int Bn = 8;
constexpr int Tn = 64;
constexpr int NI = 1024;
constexpr int NO = 1024;
constexpr int WAVES = 8;                   // waves per workgroup (256 threads)
constexpr int COLS_PER_WAVE = NI / WAVES;  // 128
constexpr int TILES = COLS_PER_WAVE / 16;  // 8 tiles of 16x16 per wave

__global__ __launch_bounds__(256) void oja_scan_kernel(
    const float* __restrict__ X,    // [B, T, NI]
    const float* __restrict__ W0,   // [B, NO, NI]
    float* __restrict__ out)        // [B, T, NO]
{
  const int lane  = threadIdx.x & 31;
  const int wv    = threadIdx.x >> 5;   // wave id 0..7
  const int b     = blockIdx.x >> 6;    // batch
  const int strip = blockIdx.x & 63;    // 16-row strip within NO
  const int half  = lane >> 4;          // 0: lanes 0-15, 1: lanes 16-31
  const int ln    = lane & 15;
  const int colbase = wv * COLS_PER_WAVE;
  const float lr = 1.0f / (float)NI;

  __shared__ float part[WAVES][16];     // per-wave dot partials per row
  __shared__ float acoef[16];           // lr*y/P_new per row

  // ---- Load V = W_init strip, native WMMA C/D layout:
  //      VGPR r, lanes 0-15 -> row r, col ln ; lanes 16-31 -> row r+8, col ln
  v8f w[TILES];
  const float* Wp = W0 + ((size_t)b * NO + (size_t)strip * 16) * NI;
#pragma unroll
  for (int j = 0; j < TILES; ++j) {
#pragma unroll
    for (int r = 0; r < 8; ++r) {
      w[j][r] = Wp[(size_t)(r + half * 8) * NI + (colbase + j * 16 + ln)];
    }
  }

  float P = 1.0f;  // running decay product (valid in wave0 lanes 0..15)

#pragma unroll 1
  for (int t = 0; t < Tn; ++t) {
    // ---- x_t slice for this wave's 128 columns (L2-resident, 2 MB total)
    const float* xp = X + ((size_t)b * Tn + t) * NI + colbase;
    float xv[TILES];
#pragma unroll
    for (int j = 0; j < TILES; ++j) xv[j] = xp[j * 16 + ln];

    // ---- partial dot products: acc[r] = sum_j V[row(r),16j+ln] * x[16j+ln]
    float acc[8];
#pragma unroll
    for (int r = 0; r < 8; ++r) acc[r] = 0.0f;
#pragma unroll
    for (int j = 0; j < TILES; ++j) {
#pragma unroll
      for (int r = 0; r < 8; ++r) acc[r] = __builtin_fmaf(w[j][r], xv[j], acc[r]);
    }

    // ---- reduce across the 16 lanes of each half (XOR swizzle tree)
#pragma unroll
    for (int r = 0; r < 8; ++r) {
      SWZ_XOR_ADD(acc[r], 0x041F);  // xor 1
      SWZ_XOR_ADD(acc[r], 0x081F);  // xor 2
      SWZ_XOR_ADD(acc[r], 0x101F);  // xor 4
      SWZ_XOR_ADD(acc[r], 0x201F);  // xor 8
    }
    if (ln == 0) {
#pragma unroll
      for (int r = 0; r < 8; ++r) part[wv][half * 8 + r] = acc[r];
    }
    __syncthreads();

    // ---- wave0 lanes 0-15: finish reduction, sigmoid, decay bookkeeping
    if (wv == 0 && lane < 16) {
      float d = 0.0f;
#pragma unroll
      for (int k = 0; k < WAVES; ++k) d += part[k][lane];
      float pre = P * d;                       // W.x = P * (V.x)
      float y   = 1.0f / (1.0f + __expf(-pre));
      out[((size_t)b * Tn + t) * NO + (size_t)strip * 16 + lane] = y;
      P *= (1.0f - lr * y * y);                // P_{t+1} = P_t * s_t
      acoef[lane] = lr * y / P;                // coef for V update
    }
    __syncthreads();

    // ---- rank-1 WMMA update: V_tile = A(16x4) * B(4x16) + V_tile
    //      A[m,0] = coef[m]  (A VGPR0 lanes 0-15 = K=0), rest zero
    //      B[0,n] = x[n]     (B VGPR0 lanes 0-15 = K=0 row), rest zero
    float ac = acoef[ln];
    v2f A;
    A.x = half ? 0.0f : ac;
    A.y = 0.0f;
#pragma unroll
    for (int j = 0; j < TILES; ++j) {
      v2f Bm;
      Bm.x = half ? 0.0f : xv[j];
      Bm.y = 0.0f;
      w[j] = __builtin_amdgcn_wmma_f32_16x16x4_f32(
          /*neg_a=*/false, A, /*neg_b=*/false, Bm,
          /*c_mod=*/(short)0, w[j], /*reuse_a=*/false, /*reuse_b=*/false);
    }
  }
}

extern "C" void kernel_launch(void* const* d_in, const int* in_sizes, int n_in,
                              void* d_out, int out_size, void* d_ws, size_t ws_size,
                              hipStream_t stream) {
  const float* X  = (const float*)d_in[0];   // [8, 64, 1024]
  const float* W0 = (const float*)d_in[1];   // [8, 1024, 1024]
  float* out      = (float*)d_out;           // [8, 64, 1024]
  (void)in_sizes; (void)n_in; (void)out_size; (void)d_ws; (void)ws_size;

  dim3 grid(Bn * (NO / 16));  // 8 batches x 64 row-strips = 512 workgroups
  dim3 block(WAVES * 32);     // 8 waves
  oja_scan_kernel<<<grid, block, 0, stream>>>(X, W0, out);
}